// SNN_Model_a_1614907703541
// MI455X (gfx1250) — compile-verified
//
#include <hip/hip_runtime.h>

// Problem constants (match reference)
#define T_STEPS 256
#define BATCH   16384
#define IN_DIM  1024
#define NCLS    10
#define NPAD    16   // pad N=10 -> 16 for the WMMA tile

typedef __attribute__((ext_vector_type(2))) float v2f;
typedef __attribute__((ext_vector_type(8))) float v8f;

// Kernel 1: xw[B,16] = x[B,1024] @ W[10,1024]^T  via V_WMMA_F32_16X16X4_F32.
// One wave computes one 16x16 (M x Npad) tile; 8 waves per block.
__global__ __launch_bounds__(256) void snn_gemm_wmma(
    const float* __restrict__ x, const float* __restrict__ W,
    float* __restrict__ xw)
{
    // W transposed+padded into LDS: wLds[k*16 + n], n>=10 zeroed.
    // 1024*16 floats = 64 KB (WGP LDS is 320 KB). Layout is bank-conflict
    // free for the B-fragment reads below.
    __shared__ float wLds[IN_DIM * NPAD];

    for (int i = threadIdx.x; i < IN_DIM * NPAD; i += 256) {
        const int k = i >> 4;
        const int n = i & 15;
        wLds[i] = (n < NCLS) ? W[n * IN_DIM + k] : 0.0f;
    }
    __syncthreads();

    const int lane = threadIdx.x & 31;
    const int wave = threadIdx.x >> 5;
    const int tile = blockIdx.x * 8 + wave;   // 16-row tile of x
    const int half = lane >> 4;               // 0: K={0,1}, 1: K={2,3}
    const int idx  = lane & 15;               // M (for A) / N (for B) index
    const int koff = half * 2;

    // A fragment source: row (tile*16+idx), two consecutive K per lane.
    const float* aPtr = x + (size_t)(tile * 16 + idx) * IN_DIM + koff;

    v8f acc = {0.f, 0.f, 0.f, 0.f, 0.f, 0.f, 0.f, 0.f};

    #pragma unroll 4
    for (int k = 0; k < IN_DIM; k += 4) {
        // A: 16x4 f32 fragment (2 VGPRs/lane)
        v2f a = *(const v2f*)(aPtr + k);
        // B: 4x16 f32 fragment, row K striped across lanes
        v2f b;
        b.x = wLds[(k + koff + 0) * NPAD + idx];
        b.y = wLds[(k + koff + 1) * NPAD + idx];
        // 8 args: (neg_a, A, neg_b, B, c_mod, C, reuse_a, reuse_b)
        acc = __builtin_amdgcn_wmma_f32_16x16x4_f32(
            false, a, false, b, (short)0, acc, false, false);
    }

    // D layout: VGPR r -> M = r + 8*half (lanes 0-15 / 16-31), N = idx.
    const int mbase = tile * 16 + half * 8;
    #pragma unroll
    for (int r = 0; r < 8; ++r) {
        xw[(size_t)(mbase + r) * NPAD + idx] = acc[r];
    }
}

// Kernel 2: LI scan. One thread per (b,c); consecutive threads write
// consecutive f32s -> fully coalesced 128B stores per wave into the
// 168 MB output stream (the bandwidth-dominant term).
__global__ __launch_bounds__(256) void snn_li_scan(
    const float* __restrict__ xw, float* __restrict__ out)
{
    const int tid = blockIdx.x * 256 + threadIdx.x;   // < BATCH*NCLS exactly
    const int b = tid / NCLS;
    const int c = tid - b * NCLS;

    const float cur  = xw[(size_t)b * NPAD + c];
    const float aMem = 1e-3f * 100.0f;        // DT * TAU_MEM_INV = 0.1
    const float dSyn = 1.0f - 1e-3f * 200.0f; // 1 - DT*TAU_SYN_INV = 0.8

    float v = 0.0f;   // V_LEAK
    float ic = 0.0f;
    float* o = out + tid;

    #pragma unroll 8
    for (int t = 0; t < T_STEPS; ++t) {
        const float ij = ic + cur;            // i_jump
        v = fmaf(aMem, ij - v, v);            // v += 0.1*((0-v) + ij)
        *o = v;
        o += BATCH * NCLS;
        ic = ij * dSyn;
    }
}

extern "C" void kernel_launch(void* const* d_in, const int* in_sizes, int n_in,
                              void* d_out, int out_size, void* d_ws, size_t ws_size,
                              hipStream_t stream) {
    const float* x = (const float*)d_in[0];   // [16384, 1024] f32
    const float* W = (const float*)d_in[1];   // [10, 1024] f32
    float* out = (float*)d_out;               // [256, 16384, 10] f32
    float* xw  = (float*)d_ws;                // scratch: [16384, 16] f32 = 1 MB

    // 1024 tiles of 16 rows, 8 waves (one tile each) per 256-thread block.
    snn_gemm_wmma<<<BATCH / (16 * 8), 256, 0, stream>>>(x, W, xw);
    // 163840 threads, one per (b, c).
    snn_li_scan<<<(BATCH * NCLS) / 256, 256, 0, stream>>>(xw, out);
}